// EQGPAgent_75788992906233
// MI455X (gfx1250) — compile-verified
//
#include <hip/hip_runtime.h>
#include <hip/hip_bf16.h>

typedef __attribute__((ext_vector_type(2))) float v2f;
typedef __attribute__((ext_vector_type(8))) float v8f;
typedef __attribute__((ext_vector_type(4))) int v4i;

#define SDIM 8
#define SADIM 10
#define NPT 256
#define KP 16
#define NN (NPT * NPT)
#define NTILE 16  // 256/16 tiles per dimension

#if __has_builtin(__builtin_amdgcn_global_load_async_to_lds_b128) && \
    __has_builtin(__builtin_amdgcn_s_wait_asynccnt)
#define USE_ASYNC_LDS 1
#else
#define USE_ASYNC_LDS 0
#endif

#define ASG __attribute__((address_space(1)))
#define ASL __attribute__((address_space(3)))

// float-element offsets into workspace
enum : size_t {
  OFF_L    = 0,                                        // [S,N,N] data_cov -> Cholesky L (in place)
  OFF_AINV = OFF_L    + (size_t)SDIM * NN,             // [S,N,N] data_cov^{-1}
  OFF_LK   = OFF_AINV + (size_t)SDIM * NN,             // [S,N]   log k(mu, x_n)
  OFF_QA   = OFF_LK   + (size_t)SDIM * NPT,            // [S,S,N] 0.5 u_a^T M_ab u_a
  OFF_QB   = OFF_QA   + (size_t)SDIM * SDIM * NPT,     // [S,S,N] 0.5 u_b^T M_ab u_b
  OFF_V    = OFF_QB   + (size_t)SDIM * SDIM * NPT,     // [S,S,N,KP] U_a * (M+M^T)
  OFF_U    = OFF_V    + (size_t)SDIM * SDIM * NPT * KP,// [S,N,KP]  w_g * nu  (padded)
  OFF_XW   = OFF_U    + (size_t)SDIM * NPT * KP,       // [S,N,KP]  w_g * x   (padded)
  OFF_XP   = OFF_XW   + (size_t)SDIM * NPT * KP,       // [N,KP]    x padded
  OFF_RQ   = OFF_XP   + (size_t)NPT * KP,              // [S,N]     sum x^2 w
  OFF_MEXP = OFF_RQ   + (size_t)SDIM * NPT,            // [S,N]     mean_det_coeff*exp(-0.5 quad)
  OFF_CCM  = OFF_MEXP + (size_t)SDIM * NPT,            // [S,N,KP]  cross_cov_mu
  OFF_BETA = OFF_CCM  + (size_t)SDIM * NPT * KP,       // [S,N]
  OFF_MSYM = OFF_BETA + (size_t)SDIM * NPT,            // [S,S,KP,KP] M_ab + M_ab^T (padded)
  OFF_BG   = OFF_MSYM + (size_t)SDIM * SDIM * KP * KP, // [S,KP,KP] (cov + diag(scales))^{-1}
  OFF_CG   = OFF_BG   + (size_t)SDIM * KP * KP,        // [S,KP,KP] cov * Bg
  OFF_T1   = OFF_CG   + (size_t)SDIM * KP * KP,        // [S,KP]    scales * (Bg mu)
  OFF_MDC  = OFF_T1   + (size_t)SDIM * KP,             // [S] mean det coeff (+pad)
  OFF_LDT  = OFF_MDC  + 32,                            // [S,S] -0.5 * logdet R
  OFF_PC   = OFF_LDT  + 64,                            // [64,16] per-strip cov partials
  OFF_PT   = OFF_PC   + (size_t)SDIM * SDIM * NTILE,   // [8,16]  per-strip trace partials
  OFF_END  = OFF_PT   + (size_t)SDIM * NTILE
};

// ---------------------------------------------------------------------------
// 10x10 Gauss-Jordan inverse with partial pivoting; returns log|det|
// ---------------------------------------------------------------------------
__device__ float gj_inv10(float A[SADIM][SADIM], float Ainv[SADIM][SADIM]) {
  float logdet = 0.f;
  for (int i = 0; i < SADIM; ++i)
    for (int j = 0; j < SADIM; ++j) Ainv[i][j] = (i == j) ? 1.f : 0.f;
  for (int c = 0; c < SADIM; ++c) {
    int piv = c;
    float mx = fabsf(A[c][c]);
    for (int r = c + 1; r < SADIM; ++r) {
      float v = fabsf(A[r][c]);
      if (v > mx) { mx = v; piv = r; }
    }
    if (piv != c) {
      for (int j = 0; j < SADIM; ++j) {
        float t = A[c][j]; A[c][j] = A[piv][j]; A[piv][j] = t;
        t = Ainv[c][j]; Ainv[c][j] = Ainv[piv][j]; Ainv[piv][j] = t;
      }
    }
    float d = A[c][c];
    logdet += logf(fabsf(d));
    float inv = 1.f / d;
    for (int j = 0; j < SADIM; ++j) { A[c][j] *= inv; Ainv[c][j] *= inv; }
    for (int r = 0; r < SADIM; ++r) {
      if (r == c) continue;
      float f = A[r][c];
      for (int j = 0; j < SADIM; ++j) { A[r][j] -= f * A[c][j]; Ainv[r][j] -= f * Ainv[c][j]; }
    }
  }
  return logdet;
}

// ---------------------------------------------------------------------------
// S1: small per-pair / per-g linear algebra (10x10)
// ---------------------------------------------------------------------------
__global__ __launch_bounds__(64) void k_setup_small(const float* __restrict__ mfull,
                                                    const float* __restrict__ cov,
                                                    const float* __restrict__ coeff,
                                                    const float* __restrict__ scales,
                                                    float* __restrict__ w) {
  int t = threadIdx.x;
  if (t >= 64) return;
  int a = t >> 3, b = t & 7;
  float wa[SADIM], wb[SADIM];
  for (int d = 0; d < SADIM; ++d) {
    wa[d] = 1.f / scales[a * SADIM + d];
    wb[d] = 1.f / scales[b * SADIM + d];
  }
  float A[SADIM][SADIM], Ai[SADIM][SADIM];
  // R_ab = cov * (wa + wb) (column scaling) + I ; logdet + inverse
  for (int i = 0; i < SADIM; ++i)
    for (int j = 0; j < SADIM; ++j)
      A[i][j] = cov[i * SADIM + j] * (wa[j] + wb[j]) + ((i == j) ? 1.f : 0.f);
  float ld = gj_inv10(A, Ai);
  w[OFF_LDT + t] = -0.5f * ld;
  // M_ab = Rinv * cov ; store Msym = M + M^T (zero padded to 16x16)
  float M[SADIM][SADIM];
  for (int i = 0; i < SADIM; ++i)
    for (int j = 0; j < SADIM; ++j) {
      float s = 0.f;
      for (int k = 0; k < SADIM; ++k) s += Ai[i][k] * cov[k * SADIM + j];
      M[i][j] = s;
    }
  float* ms = w + OFF_MSYM + (size_t)t * KP * KP;
  for (int i = 0; i < KP; ++i)
    for (int j = 0; j < KP; ++j)
      ms[i * KP + j] = (i < SADIM && j < SADIM) ? (M[i][j] + M[j][i]) : 0.f;

  if (a == 0) {  // per-g work (g == b)
    int g = b;
    // mean det coeff: det(cov * w_g + I)^{-1/2}
    for (int i = 0; i < SADIM; ++i)
      for (int j = 0; j < SADIM; ++j)
        A[i][j] = cov[i * SADIM + j] * wb[j] + ((i == j) ? 1.f : 0.f);
    float ld2 = gj_inv10(A, Ai);
    w[OFF_MDC + g] = coeff[g] * expf(-0.5f * ld2);
    // Bg = (cov + diag(scales_g))^{-1}
    for (int i = 0; i < SADIM; ++i)
      for (int j = 0; j < SADIM; ++j)
        A[i][j] = cov[i * SADIM + j] + ((i == j) ? scales[g * SADIM + i] : 0.f);
    gj_inv10(A, Ai);
    float* bg = w + OFF_BG + (size_t)g * KP * KP;
    for (int i = 0; i < KP; ++i)
      for (int j = 0; j < KP; ++j)
        bg[i * KP + j] = (i < SADIM && j < SADIM) ? Ai[i][j] : 0.f;
    // t1 = scales_g * (Bg mu)
    float* t1 = w + OFF_T1 + (size_t)g * KP;
    for (int d = 0; d < KP; ++d) {
      float s = 0.f;
      if (d < SADIM) {
        for (int j = 0; j < SADIM; ++j) s += Ai[d][j] * mfull[j];
        s *= scales[g * SADIM + d];
      }
      t1[d] = s;
    }
    // Cg = cov * Bg
    float* cg = w + OFF_CG + (size_t)g * KP * KP;
    for (int i = 0; i < KP; ++i)
      for (int j = 0; j < KP; ++j) {
        float s = 0.f;
        if (i < SADIM && j < SADIM)
          for (int k = 0; k < SADIM; ++k) s += cov[i * SADIM + k] * Ai[k][j];
        cg[i * KP + j] = s;
      }
  }
}

// ---------------------------------------------------------------------------
// S2: per (g, n) vectors: Xw, U, r, lk, mexp, ccm, Xpad
// ---------------------------------------------------------------------------
__global__ __launch_bounds__(256) void k_per_point(const float* __restrict__ mfull,
                                                   const float* __restrict__ xin,
                                                   const float* __restrict__ coeff,
                                                   const float* __restrict__ scales,
                                                   float* __restrict__ w) {
  int idx = blockIdx.x * blockDim.x + threadIdx.x;
  if (idx >= SDIM * NPT) return;
  int g = idx >> 8, n = idx & 255;
  const float* x = xin + (size_t)n * SADIM;
  float wg[SADIM], nu[SADIM];
  float rq = 0.f, nq = 0.f;
  for (int d = 0; d < SADIM; ++d) {
    wg[d] = 1.f / scales[g * SADIM + d];
    nu[d] = x[d] - mfull[d];
    rq += x[d] * x[d] * wg[d];
    nq += nu[d] * nu[d] * wg[d];
  }
  w[OFF_RQ + idx] = rq;
  w[OFF_LK + idx] = logf(coeff[g]) - 0.5f * nq;
  float* xw = w + OFF_XW + (size_t)idx * KP;
  float* uu = w + OFF_U + (size_t)idx * KP;
  for (int k = 0; k < KP; ++k) {
    xw[k] = (k < SADIM) ? x[k] * wg[k] : 0.f;
    uu[k] = (k < SADIM) ? nu[k] * wg[k] : 0.f;
  }
  if (g == 0) {
    float* xp = w + OFF_XP + (size_t)n * KP;
    for (int k = 0; k < KP; ++k) xp[k] = (k < SADIM) ? x[k] : 0.f;
  }
  // mean quad nu^T Bg nu
  const float* bg = w + OFF_BG + (size_t)g * KP * KP;
  float mq = 0.f;
  for (int i = 0; i < SADIM; ++i) {
    float s = 0.f;
    for (int j = 0; j < SADIM; ++j) s += bg[i * KP + j] * nu[j];
    mq += nu[i] * s;
  }
  w[OFF_MEXP + idx] = w[OFF_MDC + g] * expf(-0.5f * mq);
  // cross_cov_mu column
  const float* cg = w + OFF_CG + (size_t)g * KP * KP;
  const float* t1 = w + OFF_T1 + (size_t)g * KP;
  float* cc = w + OFF_CCM + (size_t)idx * KP;
  for (int d = 0; d < KP; ++d) {
    float s = 0.f;
    if (d < SADIM) {
      s = t1[d];
      for (int j = 0; j < SADIM; ++j) s += cg[d * KP + j] * x[j];
    }
    cc[d] = s;
  }
}

// ---------------------------------------------------------------------------
// S3: per (pair, n): V = u_a^T * Msym (row), qa, qb
// ---------------------------------------------------------------------------
__global__ __launch_bounds__(256) void k_pair_point(float* __restrict__ w) {
  int p = blockIdx.x, n = threadIdx.x;
  int a = p >> 3, b = p & 7;
  const float* ms = w + OFF_MSYM + (size_t)p * KP * KP;
  const float* ua = w + OFF_U + ((size_t)a * NPT + n) * KP;
  const float* ub = w + OFF_U + ((size_t)b * NPT + n) * KP;
  float* vv = w + OFF_V + ((size_t)p * NPT + n) * KP;
  float qa = 0.f;
  for (int k = 0; k < KP; ++k) {
    float s = 0.f;
    for (int d = 0; d < KP; ++d) s += ua[d] * ms[d * KP + k];
    vv[k] = s;
    qa += s * ua[k];
  }
  w[OFF_QA + (size_t)p * NPT + n] = 0.25f * qa;
  float qb = 0.f;
  for (int k = 0; k < KP; ++k) {
    float s = 0.f;
    for (int d = 0; d < KP; ++d) s += ub[d] * ms[d * KP + k];
    qb += s * ub[k];
  }
  w[OFF_QB + (size_t)p * NPT + n] = 0.25f * qb;
}

// ---------------------------------------------------------------------------
// WMMA helpers. Fragment layout per CDNA5 ISA (32-bit A 16x4): lanes 0-15 hold
// K=k0,k0+1, lanes 16-31 hold K=k0+2,k0+3. C/D: VGPR r -> M = r+8*(lane>=16).
// ---------------------------------------------------------------------------
__device__ inline void load_afrag(v2f af[4], const float* __restrict__ Arow, int lane) {
  int half = lane >> 4, lm = lane & 15;
  const float* ar = Arow + (size_t)lm * KP + 2 * half;
#pragma unroll
  for (int kc = 0; kc < 4; ++kc) { af[kc].x = ar[4 * kc]; af[kc].y = ar[4 * kc + 1]; }
}

__device__ inline v8f wmma_with_a(const v2f af[4], const float* __restrict__ br) {
  v8f c = {0.f, 0.f, 0.f, 0.f, 0.f, 0.f, 0.f, 0.f};
#pragma unroll
  for (int kc = 0; kc < 4; ++kc) {
    v2f b;
    b.x = br[4 * kc]; b.y = br[4 * kc + 1];
    c = __builtin_amdgcn_wmma_f32_16x16x4_f32(false, af[kc], false, b, (short)0, c, false, false);
  }
  return c;
}

// ---------------------------------------------------------------------------
// data_cov: one wave per (g, n-strip), loops all m-tiles; A-frag reused 16x
// quad = r_n + r_m - 2*(XW X^T)[n,m]
// ---------------------------------------------------------------------------
__global__ __launch_bounds__(32) void k_datacov(const float* __restrict__ coeff,
                                                const float* __restrict__ noise,
                                                float* __restrict__ w) {
  int g = blockIdx.y;
  int n0 = blockIdx.x * 16;
  int lane = threadIdx.x;
  int half = lane >> 4, lm = lane & 15;

  v2f af[4];
  load_afrag(af, w + OFF_XW + ((size_t)g * NPT + n0) * KP, lane);

  const float* rq = w + OFF_RQ + (size_t)g * NPT;
  float rq_n[8];
#pragma unroll
  for (int r = 0; r < 8; ++r) rq_n[r] = rq[n0 + r + 8 * half];

  float cg = coeff[g], ng = noise[g];
  float* L = w + OFF_L + (size_t)g * NN;

  for (int mt = 0; mt < NTILE; ++mt) {
    int m0 = mt * 16;
    const float* br = w + OFF_XP + ((size_t)m0 + lm) * KP + 2 * half;
    v8f c = wmma_with_a(af, br);
    int m = m0 + lm;
    float rqm = rq[m];
#pragma unroll
    for (int r = 0; r < 8; ++r) {
      int n = n0 + r + 8 * half;
      float quad = rq_n[r] + rqm - 2.f * c[r];
      float v = cg * expf(-0.5f * quad);
      if (n == m) v += ng;
      L[(size_t)n * NPT + m] = v;
    }
  }
}

// ---------------------------------------------------------------------------
// In-place Cholesky, one workgroup per g
// ---------------------------------------------------------------------------
__global__ __launch_bounds__(256) void k_chol(float* __restrict__ w) {
  int g = blockIdx.x, t = threadIdx.x;
  float* L = w + OFF_L + (size_t)g * NN;
  for (int k = 0; k < NPT; ++k) {
    if (t == 0) L[(size_t)k * NPT + k] = sqrtf(L[(size_t)k * NPT + k]);
    __syncthreads();
    float d = L[(size_t)k * NPT + k];
    int i = k + 1 + t;
    if (i < NPT) L[(size_t)i * NPT + k] /= d;
    __syncthreads();
    if (t > k) {
      float lik = L[(size_t)t * NPT + k];
      for (int j = k + 1; j <= t; ++j)
        L[(size_t)t * NPT + j] -= lik * L[(size_t)j * NPT + k];
    }
    __syncthreads();
  }
}

// ---------------------------------------------------------------------------
// A^{-1} columns via forward/backward substitution; thread = column
// ---------------------------------------------------------------------------
__global__ __launch_bounds__(256) void k_ainv(float* __restrict__ w) {
  int g = blockIdx.x, cidx = threadIdx.x;
  const float* L = w + OFF_L + (size_t)g * NN;
  float* Ai = w + OFF_AINV + (size_t)g * NN;
  float y[NPT];
  for (int i = 0; i < NPT; ++i) {
    float s = (i == cidx) ? 1.f : 0.f;
    for (int j = 0; j < i; ++j) s -= L[(size_t)i * NPT + j] * y[j];
    y[i] = s / L[(size_t)i * NPT + i];
  }
  for (int i = NPT - 1; i >= 0; --i) {
    float s = y[i];
    for (int j = i + 1; j < NPT; ++j) s -= L[(size_t)j * NPT + i] * y[j];
    y[i] = s / L[(size_t)i * NPT + i];
  }
  for (int i = 0; i < NPT; ++i) Ai[(size_t)i * NPT + cidx] = y[i];
}

// ---------------------------------------------------------------------------
// beta = A^{-1} y_g
// ---------------------------------------------------------------------------
__global__ __launch_bounds__(256) void k_beta(const float* __restrict__ yout,
                                              float* __restrict__ w) {
  int idx = blockIdx.x * blockDim.x + threadIdx.x;
  if (idx >= SDIM * NPT) return;
  int g = idx >> 8, i = idx & 255;
  const float* Ai = w + OFF_AINV + (size_t)g * NN + (size_t)i * NPT;
  float s = 0.f;
  for (int j = 0; j < NPT; ++j) s += Ai[j] * yout[j * SDIM + g];
  w[OFF_BETA + idx] = s;
}

// ---------------------------------------------------------------------------
// Q tensor: one wave per (pair, n-strip); loops all m-tiles with the A-frag
// resident. B tile staged into LDS via gfx1250 async copy when available.
// Fuses exp, beta_a Q beta_b and trace(A^{-1} Q) contractions.
// ---------------------------------------------------------------------------
__global__ __launch_bounds__(32) void k_q(float* __restrict__ w) {
  int p = blockIdx.y;
  int a = p >> 3, b = p & 7;
  int n0 = blockIdx.x * 16;
  int lane = threadIdx.x;
  int half = lane >> 4, lm = lane & 15;

  v2f af[4];
  load_afrag(af, w + OFF_V + ((size_t)p * NPT + n0) * KP, lane);

  float lk_n[8], qa_n[8], be_n[8];
  {
    const float* lka = w + OFF_LK + (size_t)a * NPT;
    const float* qan = w + OFF_QA + (size_t)p * NPT;
    const float* bea = w + OFF_BETA + (size_t)a * NPT;
#pragma unroll
    for (int r = 0; r < 8; ++r) {
      int n = n0 + r + 8 * half;
      lk_n[r] = lka[n]; qa_n[r] = qan[n]; be_n[r] = bea[n];
    }
  }

  float ldt = w[OFF_LDT + p];
  const float* Ubase = w + OFF_U + (size_t)b * NPT * KP;
  const float* lkb_v = w + OFF_LK + (size_t)b * NPT;
  const float* qb_v = w + OFF_QB + (size_t)p * NPT;
  const float* beb_v = w + OFF_BETA + (size_t)b * NPT;
  const float* Ai = w + OFF_AINV + (size_t)a * NN;

  __builtin_prefetch(Ubase + (size_t)lane * 16, 0, 1);  // global_prefetch_b8

#if USE_ASYNC_LDS
  __shared__ __align__(16) float bsm[16 * KP];
#endif

  float covacc = 0.f, tracc = 0.f;
  for (int mt = 0; mt < NTILE; ++mt) {
    int m0 = mt * 16;
#if USE_ASYNC_LDS
    {
      // Stage the 16x16 f32 B tile (1KB) into LDS: 2x b128 per lane.
      int row = lane >> 1, ch = lane & 1;
      const float* g0 = Ubase + ((size_t)m0 + row) * KP + ch * 4;
      const float* g1 = Ubase + ((size_t)m0 + row) * KP + (2 + ch) * 4;
      __builtin_amdgcn_global_load_async_to_lds_b128(
          (ASG v4i*)g0, (ASL v4i*)&bsm[row * KP + ch * 4], 0, 0);
      __builtin_amdgcn_global_load_async_to_lds_b128(
          (ASG v4i*)g1, (ASL v4i*)&bsm[row * KP + (2 + ch) * 4], 0, 0);
      __builtin_amdgcn_s_wait_asynccnt(0);
    }
    const float* br = &bsm[lm * KP + 2 * half];
#else
    const float* br = Ubase + ((size_t)m0 + lm) * KP + 2 * half;
#endif
    v8f c = wmma_with_a(af, br);

    int m = m0 + lm;
    float lkb = lkb_v[m], qbm = qb_v[m], betb = beb_v[m];
#pragma unroll
    for (int r = 0; r < 8; ++r) {
      int n = n0 + r + 8 * half;
      float q = expf(lk_n[r] + lkb + qa_n[r] + qbm + 0.5f * c[r] + ldt);
      covacc += be_n[r] * q * betb;
      if (a == b) tracc += Ai[(size_t)m * NPT + n] * q;
    }
  }

  for (int off = 16; off > 0; off >>= 1) {
    covacc += __shfl_xor(covacc, off, 32);
    tracc += __shfl_xor(tracc, off, 32);
  }
  if (lane == 0) {
    w[OFF_PC + (size_t)p * NTILE + blockIdx.x] = covacc;
    if (a == b) w[OFF_PT + (size_t)a * NTILE + blockIdx.x] = tracc;
  }
}

// ---------------------------------------------------------------------------
// Final assembly: mean [8] ++ cov [8x8] -> d_out (72 floats)
// ---------------------------------------------------------------------------
__global__ __launch_bounds__(128) void k_final(const float* __restrict__ mfull,
                                               const float* __restrict__ cov,
                                               const float* __restrict__ coeff,
                                               float* __restrict__ out,
                                               float* __restrict__ w) {
  __shared__ float sc[64], st[8], mr[8], cc[80];
  int t = threadIdx.x;
  if (t < 64) {
    float s = 0.f;
    for (int k = 0; k < NTILE; ++k) s += w[OFF_PC + (size_t)t * NTILE + k];
    sc[t] = s;
  }
  if (t < 8) {
    float s = 0.f;
    for (int k = 0; k < NTILE; ++k) s += w[OFF_PT + (size_t)t * NTILE + k];
    st[t] = s;
    float m = 0.f;
    for (int i = 0; i < NPT; ++i)
      m += w[OFF_BETA + (size_t)t * NPT + i] * w[OFF_MEXP + (size_t)t * NPT + i];
    mr[t] = m;
  }
  if (t < 80) {
    int d = t >> 3, g = t & 7;
    float s = 0.f;
    for (int i = 0; i < NPT; ++i)
      s += w[OFF_MEXP + (size_t)g * NPT + i] *
           w[OFF_CCM + ((size_t)g * NPT + i) * KP + d] *
           w[OFF_BETA + (size_t)g * NPT + i];
    cc[t] = s;
  }
  __syncthreads();
  if (t < 8) out[t] = mr[t] + mfull[t];
  if (t < 64) {
    int aa = t >> 3, bb = t & 7;
    float v = sc[t] - mr[aa] * mr[bb] + cov[aa * SADIM + bb];
    if (aa == bb) v += coeff[aa] - st[aa];
    float ccs_ab = cc[aa * 8 + bb] - mfull[aa] * mr[bb];
    float ccs_ba = cc[bb * 8 + aa] - mfull[bb] * mr[aa];
    v += ccs_ab + ccs_ba;
    out[8 + t] = v;
  }
}

extern "C" void kernel_launch(void* const* d_in, const int* in_sizes, int n_in,
                              void* d_out, int out_size, void* d_ws, size_t ws_size,
                              hipStream_t stream) {
  (void)in_sizes; (void)n_in; (void)out_size; (void)ws_size;
  const float* mfull  = (const float*)d_in[0];
  const float* cov    = (const float*)d_in[1];
  const float* xin    = (const float*)d_in[2];
  const float* yout   = (const float*)d_in[3];
  const float* coeff  = (const float*)d_in[4];
  const float* scales = (const float*)d_in[5];
  const float* noise  = (const float*)d_in[6];
  float* out = (float*)d_out;
  float* w = (float*)d_ws;

  k_setup_small<<<1, 64, 0, stream>>>(mfull, cov, coeff, scales, w);
  k_per_point<<<8, 256, 0, stream>>>(mfull, xin, coeff, scales, w);
  k_pair_point<<<64, 256, 0, stream>>>(w);
  k_datacov<<<dim3(NTILE, 8), 32, 0, stream>>>(coeff, noise, w);
  k_chol<<<8, 256, 0, stream>>>(w);
  k_ainv<<<8, 256, 0, stream>>>(w);
  k_beta<<<8, 256, 0, stream>>>(yout, w);
  k_q<<<dim3(NTILE, 64), 32, 0, stream>>>(w);
  k_final<<<1, 128, 0, stream>>>(mfull, cov, coeff, out, w);
}